// Mu_16630113370940
// MI455X (gfx1250) — compile-verified
//
#include <hip/hip_runtime.h>
#include <hip/hip_bf16.h>
#include <math.h>

typedef __attribute__((ext_vector_type(2))) float v2f;
typedef __attribute__((ext_vector_type(8))) float v8f;

#define C_DIM 256

// ---------------------------------------------------------------------------
// Kernel 1: zero the degree counters and the accumulation buffer.
// ---------------------------------------------------------------------------
__global__ void gcn_zero_kernel(unsigned* __restrict__ deg,
                                float* __restrict__ accum, int N) {
    int i = blockIdx.x * blockDim.x + threadIdx.x;
    if (i < N) {
        deg[i]   = 0u;
        accum[i] = 0.0f;
    }
}

// ---------------------------------------------------------------------------
// Kernel 2: h = x @ W via V_WMMA_F32_16X16X4_F32.
// Each wave (32 lanes) owns a 16-node tile and sweeps K=256 in steps of 4.
// A layout (32-bit A 16x4): lanes 0-15 hold K=k0,k0+1 ; lanes 16-31 hold
// K=k0+2,k0+3 -> one v2f load per lane. B broadcasts W across all 16 columns,
// so every column of the 16x16 f32 accumulator carries the same matvec value.
// ---------------------------------------------------------------------------
__global__ void gcn_matvec_wmma_kernel(const float* __restrict__ x,
                                       const float* __restrict__ W,
                                       float* __restrict__ h, int N) {
    const int lane = threadIdx.x & 31;
    const int wave = threadIdx.x >> 5;
    const int wavesPerBlock = blockDim.x >> 5;
    const int base = (blockIdx.x * wavesPerBlock + wave) * 16;   // 16 nodes/wave
    if (base >= N) return;                 // wave-uniform: EXEC stays all-1s

    const int half = lane >> 4;            // 0 = lanes 0-15, 1 = lanes 16-31
    const int m    = lane & 15;            // node-in-tile for A loads
    int row = base + m;
    if (row >= N) row = N - 1;             // clamp (duplicate loads, safe)

    const float* __restrict__ xrow = x + (size_t)row * C_DIM;

    v8f acc = {0.f, 0.f, 0.f, 0.f, 0.f, 0.f, 0.f, 0.f};

    #pragma unroll 4
    for (int k0 = 0; k0 < C_DIM; k0 += 4) {
        const int ka = k0 + half * 2;      // this lane's two K indices
        v2f a = *(const v2f*)(xrow + ka);  // A tile fragment (16x4 f32)
        v2f b;                             // B fragment: W[k] broadcast on cols
        b.x = W[ka];
        b.y = W[ka + 1];
        // D = A(16x4) * B(4x16) + C   -> v_wmma_f32_16x16x4_f32
        acc = __builtin_amdgcn_wmma_f32_16x16x4_f32(
            /*neg_a=*/false, a, /*neg_b=*/false, b,
            /*c_mod=*/(short)0, acc, /*reuse_a=*/false, /*reuse_b=*/false);
    }

    // C/D layout: lane l<16, VGPR r = D[r][l]; lane l>=16, VGPR r = D[r+8][l-16].
    // All columns identical -> lane 0 writes rows 0-7, lane 16 writes rows 8-15.
    if ((lane & 15) == 0) {
        const int mbase = base + half * 8;
        #pragma unroll
        for (int r = 0; r < 8; ++r) {
            if (mbase + r < N) h[mbase + r] = acc[r];
        }
    }
}

// ---------------------------------------------------------------------------
// Kernel 3: degree of destination nodes (unit edge weights).
// edge_index is [2, E] flattened: row 0 = src (0..E-1), row 1 = dst (E..2E-1).
// ---------------------------------------------------------------------------
__global__ void gcn_degree_kernel(const int* __restrict__ ei,
                                  unsigned* __restrict__ deg, int E) {
    int e = blockIdx.x * blockDim.x + threadIdx.x;
    if (e < E) {
        atomicAdd(&deg[ei[E + e]], 1u);
    }
}

// ---------------------------------------------------------------------------
// Kernel 4: dinv = rsqrt(deg + 1)   (+1 accounts for the self-loop; deg+1 > 0
// always, so the `where` in the reference is vacuous here).
// ---------------------------------------------------------------------------
__global__ void gcn_dinv_kernel(const unsigned* __restrict__ deg,
                                float* __restrict__ dinv, int N) {
    int i = blockIdx.x * blockDim.x + threadIdx.x;
    if (i < N) {
        dinv[i] = rsqrtf((float)(deg[i] + 1u));
    }
}

// ---------------------------------------------------------------------------
// Kernel 5: per-edge message scatter: accum[dst] += dinv[src]*dinv[dst]*h[src].
// h/dinv (~1.2 MB) sit in L2; the atomics resolve at L2 atomic units.
// ---------------------------------------------------------------------------
__global__ void gcn_edge_kernel(const int* __restrict__ ei,
                                const float* __restrict__ dinv,
                                const float* __restrict__ h,
                                float* __restrict__ accum, int E) {
    int e = blockIdx.x * blockDim.x + threadIdx.x;
    if (e < E) {
        int s = ei[e];
        int d = ei[E + e];
        atomicAdd(&accum[d], dinv[s] * dinv[d] * h[s]);
    }
}

// ---------------------------------------------------------------------------
// Kernel 6: add self-loop term + bias, apply numerically stable softplus:
// softplus(v) = max(v,0) + log1p(exp(-|v|))
// ---------------------------------------------------------------------------
__global__ void gcn_finalize_kernel(const float* __restrict__ accum,
                                    const float* __restrict__ dinv,
                                    const float* __restrict__ h,
                                    const float* __restrict__ b,
                                    float* __restrict__ out, int N) {
    int i = blockIdx.x * blockDim.x + threadIdx.x;
    if (i < N) {
        float di = dinv[i];
        float v  = accum[i] + di * di * h[i] + b[0];
        out[i]   = fmaxf(v, 0.0f) + log1pf(expf(-fabsf(v)));
    }
}

// ---------------------------------------------------------------------------
// Launcher. Inputs: d_in[0]=x [N,256] f32, d_in[1]=edge_index [2,E] i32,
// d_in[2]=W [256,1] f32, d_in[3]=b [1] f32. Output: [N,1] f32.
// Workspace layout: h[N] | deg[N] | dinv[N] | accum[N]  (4*N*4 bytes).
// ---------------------------------------------------------------------------
extern "C" void kernel_launch(void* const* d_in, const int* in_sizes, int n_in,
                              void* d_out, int out_size, void* d_ws, size_t ws_size,
                              hipStream_t stream) {
    const float* x  = (const float*)d_in[0];
    const int*   ei = (const int*)d_in[1];
    const float* W  = (const float*)d_in[2];
    const float* b  = (const float*)d_in[3];
    float* out = (float*)d_out;

    const int N = in_sizes[0] / C_DIM;   // 100000
    const int E = in_sizes[1] / 2;       // 3200000

    float*    h     = (float*)d_ws;
    unsigned* deg   = (unsigned*)(h + N);
    float*    dinv  = (float*)(deg + N);
    float*    accum = dinv + N;

    const int BLK = 256;
    const int gridN = (N + BLK - 1) / BLK;
    const int gridE = (E + BLK - 1) / BLK;
    // matvec: 8 waves/block * 16 nodes/wave = 128 nodes per block
    const int nodesPerBlock = (BLK / 32) * 16;
    const int gridM = (N + nodesPerBlock - 1) / nodesPerBlock;

    gcn_zero_kernel<<<gridN, BLK, 0, stream>>>(deg, accum, N);
    gcn_matvec_wmma_kernel<<<gridM, BLK, 0, stream>>>(x, W, h, N);
    gcn_degree_kernel<<<gridE, BLK, 0, stream>>>(ei, deg, E);
    gcn_dinv_kernel<<<gridN, BLK, 0, stream>>>(deg, dinv, N);
    gcn_edge_kernel<<<gridE, BLK, 0, stream>>>(ei, dinv, h, accum, E);
    gcn_finalize_kernel<<<gridN, BLK, 0, stream>>>(accum, dinv, h, b, out, N);
}